// Encoder_57363583205480
// MI455X (gfx1250) — compile-verified
//
#include <hip/hip_runtime.h>
#include <hip/hip_bf16.h>

typedef float v2f __attribute__((ext_vector_type(2)));
typedef float v8f __attribute__((ext_vector_type(8)));

#define NN  100000   // nodes  (= 16*6250 = 32*3125)
#define NE  3200000  // edges
#define NG  1024     // graphs
#define DIN 128
#define HD  64
#define BN_EPS 1e-5f

// ---------------------------------------------------------------- utilities
__global__ void copy_f4(const float4* __restrict__ src, float4* __restrict__ dst, int n4) {
    int i = blockIdx.x * blockDim.x + threadIdx.x;
    if (i < n4) dst[i] = src[i];
}

// zero BN-stats region (512 floats) and the pooled output x_g (NG*HD floats)
__global__ void init_zero(float* __restrict__ stats, float* __restrict__ xg) {
    int i = blockIdx.x * blockDim.x + threadIdx.x;
    if (i < 512) stats[i] = 0.0f;
    if (i < NG * HD) xg[i] = 0.0f;
}

// ------------------------------------------------- edge scatter: agg[dst] += feat[src]
// one thread handles one (edge, 4-feature group); D4 = D/4 is a power of two (32 or 16)
template <int D4_SHIFT>
__global__ __launch_bounds__(256) void scatter_add(const float* __restrict__ feat,
                                                   const int* __restrict__ src,
                                                   const int* __restrict__ dst,
                                                   float* __restrict__ agg) {
    constexpr int D4 = 1 << D4_SHIFT;
    long long tid = (long long)blockIdx.x * blockDim.x + threadIdx.x;
    constexpr long long total = (long long)NE * D4;
    if (tid >= total) return;
    int e = (int)(tid >> D4_SHIFT);
    int q = (int)(tid & (D4 - 1));
    int s = src[e], d = dst[e];
    const float4 v = ((const float4*)(feat + (long long)s * (D4 << 2)))[q];
    float* p = agg + (long long)d * (D4 << 2) + (q << 2);
    atomicAdd(p + 0, v.x);
    atomicAdd(p + 1, v.y);
    atomicAdd(p + 2, v.z);
    atomicAdd(p + 3, v.w);
}

// ------------------------------------------------- WMMA GEMM: out = relu(bn(A) @ W + b)
// A: NN x K (row major), W: K x 64, out: NN x 64.
// BN=true folds the per-K-column affine (scale/shift) into the A operand (exact for GEMM).
// Block = 256 threads = 8 waves; each wave computes one 16x16 tile via
// V_WMMA_F32_16X16X4_F32; block covers 32 rows x 64 cols -> grid.x = NN/32 (exact,
// so EXEC is all-1s as WMMA requires).
template <int K, bool BN>
__global__ __launch_bounds__(256) void gin_gemm(const float* __restrict__ A,
                                                const float* __restrict__ W,
                                                const float* __restrict__ bias,
                                                const float* __restrict__ scale,
                                                const float* __restrict__ shift,
                                                float* __restrict__ out) {
    const int w    = threadIdx.x >> 5;               // wave 0..7
    const int lane = threadIdx.x & 31;
    const int rowTile = blockIdx.x * 2 + (w >> 2);   // 16-row tile index
    const int col     = (w & 3) << 4;                // 0,16,32,48
    const int m  = lane & 15;                        // row within tile / N within tile
    const int kh = (lane >> 4) << 1;                 // K half-offset: 0 or 2

    const float* Arow = A + (long long)(rowTile * 16 + m) * K + kh;
    const float* Wcol = W + col + m + kh * HD;
    v8f c = {};
#pragma unroll
    for (int kk = 0; kk < K; kk += 4) {
        v2f a, b;
        float a0 = Arow[kk];
        float a1 = Arow[kk + 1];
        if constexpr (BN) {
            a0 = fmaf(a0, scale[kk + kh],     shift[kk + kh]);
            a1 = fmaf(a1, scale[kk + kh + 1], shift[kk + kh + 1]);
        }
        a.x = a0;
        a.y = a1;
        // B 4x16 fragment mirrors A's K striping across half-waves
        b.x = Wcol[kk * HD];
        b.y = Wcol[(kk + 1) * HD];
        c = __builtin_amdgcn_wmma_f32_16x16x4_f32(
                /*neg_a=*/false, a, /*neg_b=*/false, b,
                /*c_mod=*/(short)0, c, /*reuse_a=*/false, /*reuse_b=*/false);
    }
    // C/D layout: VGPR j -> M = j (lanes 0-15) or j+8 (lanes 16-31), N = lane&15
    const int moff = (lane >> 4) << 3;
    const float bcol = bias[col + m];
    float* orow = out + (long long)(rowTile * 16 + moff) * HD + col + m;
#pragma unroll
    for (int j = 0; j < 8; ++j) {
        orow[(long long)j * HD] = fmaxf(c[j] + bcol, 0.0f);
    }
}

// ------------------------------------------------- BatchNorm statistics
// partial per-column sum / sumsq (64 cols), LDS-reduced, atomically merged into stats[128]
__global__ __launch_bounds__(256) void bn_partial(const float* __restrict__ h,
                                                  float* __restrict__ stats) {
    __shared__ float s_sum[HD];
    __shared__ float s_sq[HD];
    const int t = threadIdx.x;
    if (t < HD) { s_sum[t] = 0.0f; s_sq[t] = 0.0f; }
    __syncthreads();
    const int c = t & (HD - 1);   // grid stride is a multiple of 64 -> column is invariant
    float lsum = 0.0f, lsq = 0.0f;
    const long long total = (long long)NN * HD;
    for (long long i = (long long)blockIdx.x * blockDim.x + t; i < total;
         i += (long long)gridDim.x * blockDim.x) {
        float v = h[i];
        lsum += v;
        lsq  = fmaf(v, v, lsq);
    }
    atomicAdd(&s_sum[c], lsum);
    atomicAdd(&s_sq[c], lsq);
    __syncthreads();
    if (t < HD) {
        atomicAdd(&stats[t],      s_sum[t]);
        atomicAdd(&stats[HD + t], s_sq[t]);
    }
}

// stats -> per-column affine:  scale = g*rsqrt(var+eps),  shift = be - mu*scale
__global__ void bn_finalize(const float* __restrict__ stats,
                            const float* __restrict__ g, const float* __restrict__ be,
                            float* __restrict__ scale, float* __restrict__ shift) {
    int c = threadIdx.x;
    if (c >= HD) return;
    const float inv_n = 1.0f / (float)NN;
    float mu  = stats[c] * inv_n;
    float var = stats[HD + c] * inv_n - mu * mu;
    float s   = g[c] * rsqrtf(var + BN_EPS);
    scale[c] = s;
    shift[c] = fmaf(-mu, s, be[c]);
}

// ------------------------------------------------- global add pooling
__global__ __launch_bounds__(256) void pool_add(const float* __restrict__ h,
                                                const int* __restrict__ batch,
                                                float* __restrict__ xg) {
    int tid = blockIdx.x * blockDim.x + threadIdx.x;
    if (tid >= NN * HD) return;
    int n = tid >> 6;
    int c = tid & (HD - 1);
    atomicAdd(&xg[batch[n] * HD + c], h[tid]);
}

// ---------------------------------------------------------------- driver
extern "C" void kernel_launch(void* const* d_in, const int* in_sizes, int n_in,
                              void* d_out, int out_size, void* d_ws, size_t ws_size,
                              hipStream_t stream) {
    const float* x     = (const float*)d_in[0];
    const int*   eidx  = (const int*)d_in[1];
    const int*   src   = eidx;
    const int*   dst   = eidx + NE;
    const int*   batch = (const int*)d_in[2];
    const float* W1_0 = (const float*)d_in[3];
    const float* b1_0 = (const float*)d_in[4];
    const float* g_0  = (const float*)d_in[5];
    const float* be_0 = (const float*)d_in[6];
    const float* W2_0 = (const float*)d_in[7];
    const float* b2_0 = (const float*)d_in[8];
    const float* W1_1 = (const float*)d_in[9];
    const float* b1_1 = (const float*)d_in[10];
    const float* g_1  = (const float*)d_in[11];
    const float* be_1 = (const float*)d_in[12];
    const float* W2_1 = (const float*)d_in[13];
    const float* b2_1 = (const float*)d_in[14];

    float* ws = (float*)d_ws;
    float* agg0   = ws;                              // NN*DIN
    float* h_mid  = agg0  + (size_t)NN * DIN;        // NN*HD
    float* h0out  = h_mid + (size_t)NN * HD;         // NN*HD
    float* agg1   = h0out + (size_t)NN * HD;         // NN*HD
    float* h1mid  = agg1  + (size_t)NN * HD;         // NN*HD
    float* stats  = h1mid + (size_t)NN * HD;         // 512 floats:
    float* stats0 = stats;                           //   [0,128)
    float* stats1 = stats + 128;                     //   [128,256)
    float* scale0 = stats + 256;
    float* shift0 = stats + 320;
    float* scale1 = stats + 384;
    float* shift1 = stats + 448;

    float* xg = (float*)d_out;                 // G x 64
    float* h_final = (float*)d_out + NG * HD;  // N x 64

    const int T = 256;
    // init: zero stats region + pooled output
    init_zero<<<(NG * HD + T - 1) / T, T, 0, stream>>>(stats, xg);

    // ---------------- layer 0 ----------------
    copy_f4<<<(NN * DIN / 4 + T - 1) / T, T, 0, stream>>>((const float4*)x, (float4*)agg0,
                                                          NN * DIN / 4);
    {
        long long tot = (long long)NE * (DIN / 4);
        scatter_add<5><<<(unsigned)((tot + T - 1) / T), T, 0, stream>>>(x, src, dst, agg0);
    }
    gin_gemm<DIN, false><<<NN / 32, T, 0, stream>>>(agg0, W1_0, b1_0, nullptr, nullptr, h_mid);
    bn_partial<<<1024, T, 0, stream>>>(h_mid, stats0);
    bn_finalize<<<1, 64, 0, stream>>>(stats0, g_0, be_0, scale0, shift0);
    gin_gemm<HD, true><<<NN / 32, T, 0, stream>>>(h_mid, W2_0, b2_0, scale0, shift0, h0out);

    // ---------------- layer 1 ----------------
    copy_f4<<<(NN * HD / 4 + T - 1) / T, T, 0, stream>>>((const float4*)h0out, (float4*)agg1,
                                                         NN * HD / 4);
    {
        long long tot = (long long)NE * (HD / 4);
        scatter_add<4><<<(unsigned)((tot + T - 1) / T), T, 0, stream>>>(h0out, src, dst, agg1);
    }
    gin_gemm<HD, false><<<NN / 32, T, 0, stream>>>(agg1, W1_1, b1_1, nullptr, nullptr, h1mid);
    bn_partial<<<1024, T, 0, stream>>>(h1mid, stats1);
    bn_finalize<<<1, 64, 0, stream>>>(stats1, g_1, be_1, scale1, shift1);
    gin_gemm<HD, true><<<NN / 32, T, 0, stream>>>(h1mid, W2_1, b2_1, scale1, shift1, h_final);

    // ---------------- pooling ----------------
    pool_add<<<(NN * HD + T - 1) / T, T, 0, stream>>>(h_final, batch, xg);
}